// IterateLeftLayer_20289425506634
// MI455X (gfx1250) — compile-verified
//
#include <hip/hip_runtime.h>
#include <cstdint>

#define CODE_LEN   1024
#define NUM_STAGES 10
#define CLIP_V     10.0f

typedef float v2f __attribute__((ext_vector_type(2)));

// sign(a)*sign(b)*min(|a|,|b|)  (bitwise xor-sign formulation)
__device__ __forceinline__ float min_sum(float a, float b) {
    uint32_t s = (__float_as_uint(a) ^ __float_as_uint(b)) & 0x80000000u;
    float m = fminf(fabsf(a), fabsf(b));
    return __uint_as_float(__float_as_uint(m) | s);
}

__device__ __forceinline__ float clipf(float x) {
    return fminf(fmaxf(x, -CLIP_V), CLIP_V);
}

// Low 32 bits of a shared-memory pointer == workgroup-relative LDS byte offset
// (HW adds LDS_BASE to the VGPR offset for async-to-LDS ops).
__device__ __forceinline__ uint32_t lds_offset(const void* p) {
    return (uint32_t)(uintptr_t)p;
}

__global__ __launch_bounds__(512) void polar_iterate_left_kernel(
    const float* __restrict__ right,      // (B, 11, 1024)
    const float* __restrict__ left,       // (B, 11, 1024)
    const float* __restrict__ lw,         // (5, 10, 2)
    const int*   __restrict__ iter_p,     // scalar
    float*       __restrict__ out)        // (B, 11, 1024)
{
    __shared__ __align__(16) float cur[CODE_LEN];          // live (unclipped) layer
    __shared__ __align__(16) float rbuf[2][CODE_LEN];      // double-buffered right layer

    const int tid = threadIdx.x;              // 0..511, one butterfly pair per thread
    const int b   = blockIdx.x;
    const int it  = *iter_p;
    const float* wbase = lw + it * NUM_STAGES * 2;

    const size_t rowBase = (size_t)b * (NUM_STAGES + 1) * CODE_LEN;
    const float* rightRow = right + rowBase;
    const float* leftRow  = left  + rowBase;
    float*       outRow   = out   + rowBase;

    // --- Prologue: async-prefetch right layer 9 into rbuf[0] (gfx1250 async DMA path)
    {
        uint32_t loff = lds_offset(&rbuf[0][tid * 2]);
        const float* g = rightRow + 9 * CODE_LEN + tid * 2;
        asm volatile("global_load_async_to_lds_b64 %0, %1, off"
                     :: "v"(loff), "v"(g) : "memory");
    }

    // --- Copy left layer 10: clipped -> out, raw -> cur (LDS)
    {
        const v2f v = *(const v2f*)(leftRow + 10 * CODE_LEN + tid * 2);
        v2f c; c.x = clipf(v.x); c.y = clipf(v.y);
        __builtin_nontemporal_store(c, (v2f*)(outRow + 10 * CODE_LEN + tid * 2));
        *(v2f*)(&cur[tid * 2]) = v;
    }

    int buf = 0;
#pragma unroll
    for (int i = NUM_STAGES - 1; i >= 0; --i) {
        // Wait for this wave's outstanding async load, then publish across the
        // workgroup (ASYNCcnt is per-wave; barrier covers the other waves'
        // slices AND the previous stage's cur[] writes).
        asm volatile("s_wait_asynccnt 0" ::: "memory");
        __syncthreads();

        // Prefetch next right layer into the other buffer. Safe only AFTER the
        // barrier: all readers of rbuf[buf^1] (previous stage) have finished.
        if (i > 0) {
            uint32_t loff = lds_offset(&rbuf[buf ^ 1][tid * 2]);
            const float* g = rightRow + (i - 1) * CODE_LEN + tid * 2;
            asm volatile("global_load_async_to_lds_b64 %0, %1, off"
                         :: "v"(loff), "v"(g) : "memory");
        }

        // Butterfly pair owned exclusively by this thread this stage:
        // p = tid with a 0 inserted at bit i; n = p + 2^i.
        const int mask = (1 << i) - 1;
        const int p = ((tid & ~mask) << 1) | (tid & mask);
        const int n = p + (1 << i);

        const float lp1 = cur[p];          // left[i+1, pos]
        const float lp0 = cur[n];          // left[i+1, neg]
        const float rp1 = rbuf[buf][p];    // right[i, pos]
        const float rp0 = rbuf[buf][n];    // right[i, neg]

        const float w0 = wbase[i * 2 + 0];
        const float w1 = wbase[i * 2 + 1];

        const float new_pos = w0 * min_sum(lp1, lp0 + rp0);
        const float new_neg = w1 * min_sum(lp1, rp1) + lp0;

        // Exclusive ownership of (p,n) -> no intra-stage race; barrier at top
        // of next iteration orders cross-stage reuse.
        cur[p] = new_pos;
        cur[n] = new_neg;

        // Output is write-once, never re-read -> non-temporal.
        __builtin_nontemporal_store(clipf(new_pos), outRow + i * CODE_LEN + p);
        __builtin_nontemporal_store(clipf(new_neg), outRow + i * CODE_LEN + n);

        buf ^= 1;
    }
}

extern "C" void kernel_launch(void* const* d_in, const int* in_sizes, int n_in,
                              void* d_out, int out_size, void* d_ws, size_t ws_size,
                              hipStream_t stream) {
    const float* right = (const float*)d_in[0];
    const float* left  = (const float*)d_in[1];
    const float* lw    = (const float*)d_in[2];
    const int*   iter  = (const int*)d_in[3];
    float* out = (float*)d_out;

    const int B = in_sizes[0] / ((NUM_STAGES + 1) * CODE_LEN);   // 4096
    polar_iterate_left_kernel<<<B, 512, 0, stream>>>(right, left, lw, iter, out);
}